// ExponentialHawkesMLE_69947837383197
// MI455X (gfx1250) — compile-verified
//
#include <hip/hip_runtime.h>

#define CHUNK 8192
#define TPB   256
#define PER   (CHUNK / TPB)   // 32 events per thread
#define EPS_  1e-8f
#define PEN_  1000.0f

typedef __attribute__((ext_vector_type(2))) float v2f;
typedef __attribute__((ext_vector_type(8))) float v8f;
typedef __attribute__((ext_vector_type(4))) int   v4i;

// ---- CDNA5 async global->LDS staging (ASYNCcnt path), guarded -------------
#if defined(__has_builtin)
#if __has_builtin(__builtin_amdgcn_global_load_async_to_lds_b128)
#define HAVE_ASYNC_LDS 1
#endif
#if __has_builtin(__builtin_amdgcn_s_wait_asynccnt)
#define HAVE_WAIT_ASYNC 1
#endif
#endif
#ifndef HAVE_ASYNC_LDS
#define HAVE_ASYNC_LDS 0
#endif
#ifndef HAVE_WAIT_ASYNC
#define HAVE_WAIT_ASYNC 0
#endif

#if HAVE_ASYNC_LDS
typedef __attribute__((address_space(1))) v4i* gp_b128;   // global int4*
typedef __attribute__((address_space(3))) v4i* lp_b128;   // LDS int4*
#endif

__device__ __forceinline__ void stage_b128(const float* __restrict__ gsrc,
                                           float* lds_dst) {
#if HAVE_ASYNC_LDS
    __builtin_amdgcn_global_load_async_to_lds_b128(
        (gp_b128)gsrc, (lp_b128)lds_dst, 0, 0);
#else
    *(float4*)lds_dst = *(const float4*)gsrc;
#endif
}

__device__ __forceinline__ void stage_wait() {
#if HAVE_ASYNC_LDS
#if HAVE_WAIT_ASYNC
    __builtin_amdgcn_s_wait_asynccnt(0);
#else
    asm volatile("s_wait_asynccnt 0x0" ::: "memory");
#endif
#endif
}

__device__ __forceinline__ float sp_(float x) {
    // softplus, stable
    return (x > 20.0f) ? x : log1pf(__expf(x));
}

// Wave32 sum-reduction on the matrix unit.
// A (16x4 f32): lane L, VGPR0 = A[L%16][L<16?0:2] = partial, VGPR1 = 0.
// B (4x16) = all ones  -> layout independent. D[m][n] = p[m] + p[m+16].
// lane<16 sums rows 0..7, lane>=16 rows 8..15; xor-16 add => full wave sum.
__device__ __forceinline__ float wave_sum_wmma(float v) {
    v2f a; a[0] = v;    a[1] = 0.0f;
    v2f b; b[0] = 1.0f; b[1] = 1.0f;
    v8f c = {};
    c = __builtin_amdgcn_wmma_f32_16x16x4_f32(false, a, false, b,
                                              (short)0, c, false, false);
    float s = c[0] + c[1] + c[2] + c[3] + c[4] + c[5] + c[6] + c[7];
    s += __shfl_xor(s, 16, 32);
    return s;
}

// ---------------- Pass 1: per-thread + per-block affine composites --------
// map R -> A*R + B over the block's 8192 transitions.
__global__ __launch_bounds__(TPB)
void hawkes_pass1(const float* __restrict__ et, const float* __restrict__ raw_beta,
                  float2* __restrict__ threadAB, float2* __restrict__ blockAB,
                  int n) {
    __shared__ __align__(16) float sT[CHUNK + 4];
    __shared__ float sA[TPB], sB[TPB];
    const int tid = threadIdx.x;
    const int b   = blockIdx.x;
    const long base = (long)b * CHUNK;

    // stage chunk (+1 boundary element) into LDS
    if (base + CHUNK <= (long)n) {
        #pragma unroll
        for (int k = 0; k < CHUNK / 4 / TPB; ++k) {
            const int e = 4 * (tid + k * TPB);
            stage_b128(et + base + e, &sT[e]);
        }
        if (tid == 0) {
            long idx = base + CHUNK;
            sT[CHUNK] = (idx < (long)n) ? et[idx] : 0.0f;
        }
        stage_wait();
    } else {
        for (int k = tid; k <= CHUNK; k += TPB) {
            long idx = base + k;
            sT[k] = (idx < (long)n) ? et[idx] : 0.0f;
        }
    }
    const float beta = sp_(raw_beta[0]) + EPS_;
    __syncthreads();

    // serial composite over this thread's 32 transitions (R[p] -> R[p+32])
    const int p = tid * PER;
    float A = 1.0f, B = 0.0f;
    float tprev = sT[p];
    #pragma unroll
    for (int k = 1; k <= PER; ++k) {
        long i = base + p + k;
        if (i < (long)n) {
            float t = sT[p + k];
            float a = __expf(-beta * (t - tprev));
            A = a * A;
            B = a * B + a;
            tprev = t;
        }
    }
    threadAB[(long)b * TPB + tid] = make_float2(A, B);

    // inclusive composition scan over the 256 thread maps (Hillis-Steele)
    sA[tid] = A; sB[tid] = B;
    __syncthreads();
    for (int off = 1; off < TPB; off <<= 1) {
        float a1 = sA[tid], b1 = sB[tid];
        float a0 = 1.0f, b0 = 0.0f;
        if (tid >= off) { a0 = sA[tid - off]; b0 = sB[tid - off]; }
        __syncthreads();
        sA[tid] = a1 * a0;
        sB[tid] = a1 * b0 + b1;
        __syncthreads();
    }
    if (tid == 0) blockAB[b] = make_float2(sA[TPB - 1], sB[TPB - 1]);
}

// ---------------- Pass 2: scan 1024 block composites -> carry R per block --
__global__ __launch_bounds__(1024)
void hawkes_pass2(const float2* __restrict__ blockAB, float* __restrict__ carryR,
                  int nb) {
    __shared__ float sA[1024], sB[1024];
    const int tid = threadIdx.x;
    float A = 1.0f, B = 0.0f;
    if (tid < nb) { float2 c = blockAB[tid]; A = c.x; B = c.y; }
    sA[tid] = A; sB[tid] = B;
    __syncthreads();
    for (int off = 1; off < 1024; off <<= 1) {
        float a1 = sA[tid], b1 = sB[tid];
        float a0 = 1.0f, b0 = 0.0f;
        if (tid >= off) { a0 = sA[tid - off]; b0 = sB[tid - off]; }
        __syncthreads();
        sA[tid] = a1 * a0;
        sB[tid] = a1 * b0 + b1;
        __syncthreads();
    }
    // R[0] = 0  =>  incoming R for block b is B of blocks [0..b-1]
    if (tid < nb) carryR[tid] = (tid == 0) ? 0.0f : sB[tid - 1];
}

// ---------------- Pass 3: apply scan, accumulate log & integral terms -----
__global__ __launch_bounds__(TPB)
void hawkes_pass3(const float* __restrict__ et,
                  const float* __restrict__ raw_mu, const float* __restrict__ raw_alpha,
                  const float* __restrict__ raw_beta,
                  const float2* __restrict__ threadAB, const float* __restrict__ carryR,
                  float* __restrict__ blockLog, float* __restrict__ blockInt,
                  int n) {
    __shared__ __align__(16) float sT[CHUNK];
    __shared__ float sA[TPB], sB[TPB];
    __shared__ float wsum[16];
    const int tid = threadIdx.x;
    const int b   = blockIdx.x;
    const long base = (long)b * CHUNK;

    // stage chunk (async path overlaps with scalar param math below)
    if (base + CHUNK <= (long)n) {
        #pragma unroll
        for (int k = 0; k < CHUNK / 4 / TPB; ++k) {
            const int e = 4 * (tid + k * TPB);
            stage_b128(et + base + e, &sT[e]);
        }
    } else {
        for (int k = tid; k < CHUNK; k += TPB) {
            long idx = base + k;
            sT[k] = (idx < (long)n) ? et[idx] : 0.0f;
        }
    }

    const float mu    = sp_(raw_mu[0])    + EPS_;
    const float alpha = sp_(raw_alpha[0]) + EPS_;
    const float beta  = sp_(raw_beta[0])  + EPS_;
    const float T     = et[n - 1];

    // intra-block exclusive composition scan over stored thread maps
    float2 comp = threadAB[(long)b * TPB + tid];
    sA[tid] = comp.x; sB[tid] = comp.y;
    stage_wait();
    __syncthreads();
    for (int off = 1; off < TPB; off <<= 1) {
        float a1 = sA[tid], b1 = sB[tid];
        float a0 = 1.0f, b0 = 0.0f;
        if (tid >= off) { a0 = sA[tid - off]; b0 = sB[tid - off]; }
        __syncthreads();
        sA[tid] = a1 * a0;
        sB[tid] = a1 * b0 + b1;
        __syncthreads();
    }
    const float Ae = (tid == 0) ? 1.0f : sA[tid - 1];
    const float Be = (tid == 0) ? 0.0f : sB[tid - 1];

    float R = Ae * carryR[b] + Be;   // R at this thread's first event
    float lsum = 0.0f, isum = 0.0f;
    const int p = tid * PER;
    #pragma unroll
    for (int k = 0; k < PER; ++k) {
        long i = base + p + k;
        if (i < (long)n) {
            float t = sT[p + k];
            float lam = mu + alpha * R;
            lsum += __logf(lam);
            isum += 1.0f - __expf(-beta * (T - t));
            if (k < PER - 1) {
                float tn = sT[p + k + 1];
                R = __expf(-beta * (tn - t)) * (1.0f + R);
            }
        }
    }

    // wave-level reduction on the matrix unit (fully convergent here)
    float wl = wave_sum_wmma(lsum);
    float wi = wave_sum_wmma(isum);
    const int wave = tid >> 5;
    if ((tid & 31) == 0) { wsum[wave] = wl; wsum[8 + wave] = wi; }
    __syncthreads();
    if (tid == 0) {
        float L = 0.0f, I = 0.0f;
        #pragma unroll
        for (int w = 0; w < 8; ++w) { L += wsum[w]; I += wsum[8 + w]; }
        blockLog[b] = L;
        blockInt[b] = I;
    }
}

// ---------------- Final combine -------------------------------------------
__global__ __launch_bounds__(256)
void hawkes_final(const float* __restrict__ blockLog, const float* __restrict__ blockInt,
                  const float* __restrict__ et,
                  const float* __restrict__ raw_mu, const float* __restrict__ raw_alpha,
                  const float* __restrict__ raw_beta,
                  float* __restrict__ out, int n, int nb) {
    __shared__ float sL[256], sI[256];
    const int tid = threadIdx.x;
    float L = 0.0f, I = 0.0f;
    for (int i = tid; i < nb; i += 256) { L += blockLog[i]; I += blockInt[i]; }
    sL[tid] = L; sI[tid] = I;
    __syncthreads();
    for (int s = 128; s > 0; s >>= 1) {
        if (tid < s) { sL[tid] += sL[tid + s]; sI[tid] += sI[tid + s]; }
        __syncthreads();
    }
    if (tid == 0) {
        float mu    = sp_(raw_mu[0])    + EPS_;
        float alpha = sp_(raw_alpha[0]) + EPS_;
        float beta  = sp_(raw_beta[0])  + EPS_;
        float T = et[n - 1];
        float integral = mu * T + (alpha / beta) * sI[0];
        float loglik = sL[0] - integral;
        float br = alpha / beta - 0.999f;
        float r = br > 0.0f ? br : 0.0f;
        loglik -= PEN_ * r * r;
        out[0] = -loglik;
    }
}

extern "C" void kernel_launch(void* const* d_in, const int* in_sizes, int n_in,
                              void* d_out, int out_size, void* d_ws, size_t ws_size,
                              hipStream_t stream) {
    const float* et  = (const float*)d_in[0];
    const float* rmu = (const float*)d_in[1];
    const float* ral = (const float*)d_in[2];
    const float* rbe = (const float*)d_in[3];
    const int n  = in_sizes[0];
    const int nb = (n + CHUNK - 1) / CHUNK;   // 1024 for N = 2^23

    float* ws = (float*)d_ws;
    float2* threadAB = (float2*)ws;                       // nb*TPB float2
    float2* blockAB  = (float2*)(ws + 2L * nb * TPB);     // nb float2
    float*  carryR   = ws + 2L * nb * TPB + 2L * nb;      // nb
    float*  blockLog = carryR + nb;                       // nb
    float*  blockInt = blockLog + nb;                     // nb

    hawkes_pass1<<<nb, TPB, 0, stream>>>(et, rbe, threadAB, blockAB, n);
    hawkes_pass2<<<1, 1024, 0, stream>>>(blockAB, carryR, nb);
    hawkes_pass3<<<nb, TPB, 0, stream>>>(et, rmu, ral, rbe, threadAB, carryR,
                                         blockLog, blockInt, n);
    hawkes_final<<<1, 256, 0, stream>>>(blockLog, blockInt, et, rmu, ral, rbe,
                                        (float*)d_out, n, nb);
}